// EdgeguidedRankingLoss_19791209300510
// MI455X (gfx1250) — compile-verified
//
#include <hip/hip_runtime.h>
#include <math.h>

#define N_IMG 8
#define H_DIM 768
#define W_DIM 1024
#define S_DIM 30000
#define HW (H_DIM * W_DIM)           // 786432
#define NWORDS (HW / 32)             // 24576 mask words per image
#define SBLK 256
#define NSB ((S_DIM + SBLK - 1) / SBLK)  // 118 sample blocks per image
#define PART 128                         // padded partial slots (>= NSB)
#define MASKV (-1e-8f)
#define ALPHA_C 1.0f

typedef float v2f __attribute__((ext_vector_type(2)));
typedef float v8f __attribute__((ext_vector_type(8)));

// Sobel (cross-correlation, matching lax.conv) on channel-0 plane g, interior only.
__device__ __forceinline__ void sobel_gxgy(const float* __restrict__ g, int y, int x,
                                           float& gx, float& gy) {
    const float* r0 = g + (size_t)(y - 1) * W_DIM + x;
    const float* r1 = g + (size_t)y * W_DIM + x;
    const float* r2 = g + (size_t)(y + 1) * W_DIM + x;
    float a00 = r0[-1], a01 = r0[0], a02 = r0[1];
    float a10 = r1[-1],               a12 = r1[1];
    float a20 = r2[-1], a21 = r2[0], a22 = r2[1];
    gx = (a02 - a00) + 2.0f * (a12 - a10) + (a22 - a20);
    gy = (a00 + 2.0f * a01 + a02) - (a20 + 2.0f * a21 + a22);
}

__global__ void k_init(unsigned* emax, int* counts, float* parts) {
    int i = blockIdx.x * blockDim.x + threadIdx.x;
    if (i < N_IMG) { emax[i] = 0u; counts[i] = 0; }
    for (int j = i; j < 4 * N_IMG * PART; j += blockDim.x * gridDim.x) parts[j] = 0.0f;
}

__global__ void k_emax(const float* __restrict__ images, unsigned* emax) {
    int n = blockIdx.y;
    int p = blockIdx.x * blockDim.x + threadIdx.x;
    const float* g = images + (size_t)n * 3 * HW;
    float e = 0.0f;
    int y = p / W_DIM, x = p % W_DIM;
    if (y > 0 && y < H_DIM - 1 && x > 0 && x < W_DIM - 1) {
        float gx, gy; sobel_gxgy(g, y, x, gx, gy);
        e = sqrtf(gx * gx + gy * gy + 1e-6f);
    }
    __shared__ float sh[256];
    sh[threadIdx.x] = e;
    __syncthreads();
    for (int s2 = 128; s2 > 0; s2 >>= 1) {
        if (threadIdx.x < s2) sh[threadIdx.x] = fmaxf(sh[threadIdx.x], sh[threadIdx.x + s2]);
        __syncthreads();
    }
    if (threadIdx.x == 0) atomicMax(&emax[n], __float_as_uint(sh[0]));  // e >= 0: bit order == float order
}

// One wave32 per 32-pixel mask word: recompute Sobel, ballot -> packed emask.
__global__ void k_pack(const float* __restrict__ images, const unsigned* __restrict__ emax,
                       unsigned* __restrict__ words) {
    int tid = threadIdx.x;
    int lane = tid & 31;
    int gw = blockIdx.x * (blockDim.x >> 5) + (tid >> 5);
    int n = gw / NWORDS, wi = gw % NWORDS;
    const float* g = images + (size_t)n * 3 * HW;
    float thr = 0.1f * __uint_as_float(emax[n]);
    int p = wi * 32 + lane;
    int y = p / W_DIM, x = p % W_DIM;
    float e = 0.0f;
    if (y > 0 && y < H_DIM - 1 && x > 0 && x < W_DIM - 1) {
        float gx, gy; sobel_gxgy(g, y, x, gx, gy);
        e = sqrtf(gx * gx + gy * gy + 1e-6f);
    }
    unsigned long long bal = __ballot(e >= thr);
    if (lane == 0) words[gw] = (unsigned)(bal & 0xffffffffull);
}

// Per-image exclusive prefix sum over word popcounts (single block, chunked LDS scan).
__global__ void k_scan(const unsigned* __restrict__ words, int* __restrict__ prefix,
                       int* __restrict__ counts) {
    int n = blockIdx.x, tid = threadIdx.x;
    const unsigned* w = words + (size_t)n * NWORDS;
    int* pf = prefix + (size_t)n * (NWORDS + 1);
    __shared__ int sh[256];
    int running = 0;
    for (int c = 0; c < NWORDS / 256; ++c) {
        int i = c * 256 + tid;
        int v = __popc(w[i]);
        sh[tid] = v;
        __syncthreads();
        for (int off = 1; off < 256; off <<= 1) {
            int t = (tid >= off) ? sh[tid - off] : 0;
            __syncthreads();
            sh[tid] += t;
            __syncthreads();
        }
        pf[i] = running + sh[tid] - v;   // exclusive
        int ct = sh[255];
        __syncthreads();
        running += ct;
    }
    if (tid == 0) { pf[NWORDS] = running; counts[n] = running; }
}

__global__ void k_sample(const float* __restrict__ inputs, const float* __restrict__ targets,
                         const float* __restrict__ images,
                         const int* __restrict__ anchor, const int* __restrict__ distr,
                         const unsigned* __restrict__ words, const int* __restrict__ prefix,
                         const int* __restrict__ counts, float* __restrict__ parts) {
    int n = blockIdx.y;
    int b = blockIdx.x;
    int tid = threadIdx.x;
    int s = b * SBLK + tid;
    float le = 0.0f, lu = 0.0f, ce = 0.0f, cu = 0.0f;
    if (s < S_DIM) {
        int cnt = counts[n];
        int minlen = cnt > 0 ? cnt : 1;
        int r = anchor[(size_t)n * S_DIM + s] % minlen;
        int sel;
        if (cnt <= 0) {
            sel = 0;  // stable argsort of all-false mask -> identity order
        } else {
            const int* pf = prefix + (size_t)n * (NWORDS + 1);
            int lo = 0, hi = NWORDS - 1;
            while (lo < hi) {                 // largest j with pf[j] <= r
                int mid = (lo + hi + 1) >> 1;
                if (pf[mid] <= r) lo = mid; else hi = mid - 1;
            }
            int t = r - pf[lo];
            unsigned wb = words[(size_t)n * NWORDS + lo];
            for (int i = 0; i < t; ++i) wb &= wb - 1u;   // drop t lowest set bits
            sel = lo * 32 + (__ffs(wb) - 1);
        }
        int ya = sel / W_DIM, xa = sel % W_DIM;
        float theta = 0.0f;   // padded border theta == 0
        const float* g = images + (size_t)n * 3 * HW;
        if (ya > 0 && ya < H_DIM - 1 && xa > 0 && xa < W_DIM - 1) {
            float gx, gy; sobel_gxgy(g, ya, xa, gx, gy);
            theta = atan2f(gy, gx);
        }
        float st, ct; sincosf(theta, &st, &ct);
        int lin[4];
        for (int k = 0; k < 4; ++k) {
            float sg = (k < 2) ? -1.0f : 1.0f;
            float dist = ((float)distr[((size_t)n * 4 + k) * S_DIM + s] + 2.0f) * sg;
            int cc = xa + (int)rintf(dist * ct);   // RNE == jnp.round
            int rr = ya + (int)rintf(dist * st);
            cc = cc < 0 ? 0 : (cc > W_DIM - 1 ? W_DIM - 1 : cc);
            rr = rr < 0 ? 0 : (rr > H_DIM - 1 ? H_DIM - 1 : rr);
            lin[k] = rr * W_DIM + cc;
        }
        const float* inp = inputs + (size_t)n * HW;
        const float* tg  = targets + (size_t)n * HW;
        const float UP = logf(1.03f);
        const float LOv = logf(0.97087378640776699f);   // log(1/1.03) in f32
        for (int p = 0; p < 3; ++p) {
            int A = lin[p], B = lin[p + 1];
            float iA = inp[A], iB = inp[B];
            float trA = tg[A], trB = tg[B];
            float cm = ((trA > MASKV) ? 1.0f : 0.0f) * ((trB > MASKV) ? 1.0f : 0.0f);
            float lr = logf(fmaxf(trA, 1e-8f)) - logf(fmaxf(trB, 1e-8f));
            bool meq = (lr < UP) && (lr > LOv);
            float diff = iA - iB;
            if (meq) {
                float ad = fabsf(diff);
                float sl1 = (ad < 1.0f) ? 0.5f * diff * diff : ad - 0.5f;
                le += sl1 * cm;
                ce += cm;
            } else {
                float lab = (lr >= UP) ? 1.0f : ((lr <= LOv) ? -1.0f : 0.0f);
                float xx = -diff * lab;
                float sp = fmaxf(xx, 0.0f) + log1pf(expf(-fabsf(xx)));  // stable softplus
                lu += sp * cm;
                cu += cm;
            }
        }
    }
    __shared__ float sh[4][SBLK];
    sh[0][tid] = le; sh[1][tid] = lu; sh[2][tid] = ce; sh[3][tid] = cu;
    __syncthreads();
    for (int s2 = SBLK / 2; s2 > 0; s2 >>= 1) {
        if (tid < s2)
            for (int q = 0; q < 4; ++q) sh[q][tid] += sh[q][tid + s2];
        __syncthreads();
    }
    if (tid == 0)
        for (int q = 0; q < 4; ++q)
            parts[((size_t)q * N_IMG + n) * PART + b] = sh[q][0];
}

// One wave. Reduce each 128-partial stream with V_WMMA_F32_16X16X4_F32:
// D += A(16x4) x ones(4x16). B == all-ones makes the B layout irrelevant and
// A only needs to be a bijection onto the 64 matrix slots, so the sum is exact.
__global__ void k_final(const float* __restrict__ parts, float* __restrict__ out) {
    int lane = threadIdx.x;   // 0..31, EXEC all ones (required by WMMA)
    float total = 0.0f;
    for (int n = 0; n < N_IMG; ++n) {
        float sums[4];
        for (int q = 0; q < 4; ++q) {
            const float* src = parts + ((size_t)q * N_IMG + n) * PART;
            v8f c = {0.f, 0.f, 0.f, 0.f, 0.f, 0.f, 0.f, 0.f};
            v2f ones; ones.x = 1.0f; ones.y = 1.0f;
            for (int it = 0; it < PART / 64; ++it) {
                v2f a;
                a.x = src[it * 64 + lane * 2];
                a.y = src[it * 64 + lane * 2 + 1];
                c = __builtin_amdgcn_wmma_f32_16x16x4_f32(false, a, false, ones,
                                                          (short)0, c, false, false);
            }
            // D col 0: rows 0..7 live in lane 0, rows 8..15 in lane 16.
            float sloc = c[0] + c[1] + c[2] + c[3] + c[4] + c[5] + c[6] + c[7];
            sums[q] = __shfl(sloc, 0, 32) + __shfl(sloc, 16, 32);
        }
        total += ALPHA_C * sums[0] / fmaxf(sums[2], 1.0f) + sums[1] / fmaxf(sums[3], 1.0f);
    }
    if (lane == 0) out[0] = total / (float)N_IMG;
}

extern "C" void kernel_launch(void* const* d_in, const int* in_sizes, int n_in,
                              void* d_out, int out_size, void* d_ws, size_t ws_size,
                              hipStream_t stream) {
    const float* inputs  = (const float*)d_in[0];
    const float* targets = (const float*)d_in[1];
    const float* images  = (const float*)d_in[2];
    const int*   anchor  = (const int*)d_in[3];
    const int*   distr   = (const int*)d_in[4];
    float* out = (float*)d_out;

    // Workspace layout (all 4-byte units): ~1.6 MB total.
    unsigned* ws_u  = (unsigned*)d_ws;
    unsigned* emax  = ws_u;                                   // [8]
    int*      counts = (int*)(ws_u + 8);                      // [8]
    unsigned* words = ws_u + 16;                              // [N*NWORDS]
    int*      prefix = (int*)(ws_u + 16 + N_IMG * NWORDS);    // [N*(NWORDS+1)]
    float*    parts  = (float*)(ws_u + 16 + N_IMG * NWORDS + N_IMG * (NWORDS + 1)); // [4*N*PART]

    k_init<<<16, 256, 0, stream>>>(emax, counts, parts);
    k_emax<<<dim3(HW / 256, N_IMG), 256, 0, stream>>>(images, emax);
    k_pack<<<(N_IMG * NWORDS) / 8, 256, 0, stream>>>(images, emax, words);
    k_scan<<<N_IMG, 256, 0, stream>>>(words, prefix, counts);
    k_sample<<<dim3(NSB, N_IMG), SBLK, 0, stream>>>(inputs, targets, images, anchor, distr,
                                                    words, prefix, counts, parts);
    k_final<<<1, 32, 0, stream>>>(parts, out);
}